// MouthLoss_14439680049444
// MI455X (gfx1250) — compile-verified
//
#include <hip/hip_runtime.h>
#include <stdint.h>

// ---------------------------------------------------------------------------
// MouthLoss on MI455X (gfx1250).
// Tiny problem (~1 MB inputs, ~0.3 GFLOP) -> L2/LDS-resident, latency bound.
// No matmul structure => WMMA not applicable; use CDNA5 async-to-LDS path
// (ASYNCcnt) + 320KB LDS to keep the full edge table resident per workgroup.
// Edge table packed as float2/float4 so hot loops issue ds_load_b64/b128.
// ---------------------------------------------------------------------------

#define V_N 65536
#define E_N 4096
#define P_N 4096
#define T_N 512
#define SENTF 1000000000.0f
#define BIGF  1e30f
#define FLTMAXF 3.402823466e+38f

#define TPB 256           // 8 waves of 32
#define PTS_PER_BLOCK 32  // 8 waves x 4 points
#define PTS_PER_WAVE 4
#define SCAN_TPB 256

// ---- CDNA5 async global->LDS copies (guarded; sync fallback keeps build OK)
#if defined(__has_builtin)
#if __has_builtin(__builtin_amdgcn_global_load_async_to_lds_b128) && \
    __has_builtin(__builtin_amdgcn_global_load_async_to_lds_b32)  && \
    __has_builtin(__builtin_amdgcn_s_wait_asynccnt)
#define HAVE_ASYNC_LDS 1
#endif
#endif

// Exact pointer types per hipcc diagnostics:
//  b128: (int __vector(4) AS1*, int __vector(4) AS3*, imm, imm)
//  b32 : (int AS1*, int AS3*, imm, imm)
typedef int v4i_t __attribute__((vector_size(4 * sizeof(int))));
typedef __attribute__((address_space(1))) v4i_t* g128_t;
typedef __attribute__((address_space(3))) v4i_t* l128_t;
typedef __attribute__((address_space(1))) int*   g32_t;
typedef __attribute__((address_space(3))) int*   l32_t;

__device__ __forceinline__ void async_copy_b128(const float* g, float* l) {
#if defined(HAVE_ASYNC_LDS)
  __builtin_amdgcn_global_load_async_to_lds_b128((g128_t)g, (l128_t)l, 0, 0);
#else
  *(float4*)l = *(const float4*)g;
#endif
}

__device__ __forceinline__ void async_copy_b32(const float* g, float* l) {
#if defined(HAVE_ASYNC_LDS)
  __builtin_amdgcn_global_load_async_to_lds_b32((g32_t)g, (l32_t)l, 0, 0);
#else
  *l = *g;
#endif
}

__device__ __forceinline__ void async_wait_all() {
#if defined(HAVE_ASYNC_LDS)
  __builtin_amdgcn_s_wait_asynccnt(0);
#endif
}

// ---------------------------------------------------------------------------
// Workspace layout (floats):
//   ep[0 .. 2E)   : (xmn, xmx) pairs,   float2 per edge
//   ep[2E .. 6E)  : (a, b, ymn, ymx),   float4 per edge
//   then m[P] floats, valid[P] ints
// ---------------------------------------------------------------------------

// Kernel 1: per-edge params, packed for b64/b128 LDS reads.
__global__ void edge_params_kernel(const float* __restrict__ verts,
                                   const int*   __restrict__ edges,
                                   float* __restrict__ ep) {
  int e = blockIdx.x * blockDim.x + threadIdx.x;
  if (e >= E_N) return;
  int i0 = edges[2 * e];
  int i1 = edges[2 * e + 1];
  float x0 = verts[3 * i0], y0 = verts[3 * i0 + 1];
  float x1 = verts[3 * i1], y1 = verts[3 * i1 + 1];
  float a = (y0 - y1) / (x0 - x1);
  float b = y0 - a * x0;
  float2 pr;
  pr.x = fminf(x0, x1);
  pr.y = fmaxf(x0, x1);
  ((float2*)ep)[e] = pr;
  float4 q;
  q.x = a;
  q.y = b;
  q.z = fminf(y0, y1);
  q.w = fmaxf(y0, y1);
  ((float4*)(ep + 2 * E_N))[e] = q;
}

// ---------------------------------------------------------------------------
// Kernel 2: per-point scans (one wave32 per point), edge table in LDS.
// ---------------------------------------------------------------------------
__global__ void __launch_bounds__(TPB)
point_kernel(const float* __restrict__ verts,
             const float* __restrict__ tableR,
             const int*   __restrict__ listAll,
             const float* __restrict__ ep,
             float* __restrict__ mOut,
             int*   __restrict__ vOut) {
  // 32KB pairs + 64KB quads + 8KB table = 104KB (<= 320KB/WGP, 3 wgs/WGP)
  __shared__ __align__(16) float sPair[2 * E_N];   // (xmn,xmx) per edge
  __shared__ __align__(16) float sQuad[4 * E_N];   // (a,b,ymn,ymx) per edge
  __shared__ __align__(16) float sT[4 * T_N];      // (c3,c4,c5,c6) per row

  const int tid = threadIdx.x;

  // Stage edge tables: contiguous b128 async transfers.
  for (int i = tid; i < (2 * E_N) / 4; i += TPB)
    async_copy_b128(ep + 4 * i, sPair + 4 * i);
  for (int i = tid; i < (4 * E_N) / 4; i += TPB)
    async_copy_b128(ep + 2 * E_N + 4 * i, sQuad + 4 * i);
  // Stage TableR columns 3..6 interleaved as float4 rows (stride-7 source,
  // 4B-aligned -> element-wise b32 async).
  for (int t = tid; t < T_N; t += TPB) {
    async_copy_b32(tableR + 7 * t + 3, sT + 4 * t + 0);
    async_copy_b32(tableR + 7 * t + 4, sT + 4 * t + 1);
    async_copy_b32(tableR + 7 * t + 5, sT + 4 * t + 2);
    async_copy_b32(tableR + 7 * t + 6, sT + 4 * t + 3);
  }
  async_wait_all();
  __syncthreads();

  const float2* sPr = (const float2*)sPair;
  const float4* sQd = (const float4*)sQuad;
  const float4* sTr = (const float4*)sT;

  const int lane = tid & 31;
  const int wave = tid >> 5;

  for (int k = 0; k < PTS_PER_WAVE; ++k) {
    const int pt = blockIdx.x * PTS_PER_BLOCK + wave * PTS_PER_WAVE + k;
    const int vidx = listAll[pt];
    const float px = verts[3 * vidx];
    const float py = verts[3 * vidx + 1];

    // ---- pass 1: first edge index where px in (xmn, xmx); else E-1 --------
    int best = E_N;
    for (int e = lane; e < E_N; e += 32) {
      float2 mm = sPr[e];                       // ds_load_b64, conflict-free
      if (px > mm.x && px < mm.y) { best = e; break; }  // e increasing per lane
    }
#pragma unroll
    for (int off = 16; off; off >>= 1) {
      int o = __shfl_xor(best, off, 32);
      best = (o < best) ? o : best;
    }
    const int idx = (best == E_N) ? (E_N - 1) : best;
    const float aI = sQuad[4 * idx];
    const float bI = sQuad[4 * idx + 1];
    const float exposeY = aI * px + bI;
    const float L1 = fabsf(py - exposeY);
    const float cy = 0.5f * (py + exposeY);
    const float cx = px;

    // ---- pass 2: condB reductions ----------------------------------------
    int   n = 0, anyS = 0, anyI = 0;
    float xallmx = -SENTF, xallmn = SENTF;
    float xs = SENTF, xinf = -SENTF;
    for (int e = lane; e < E_N; e += 32) {
      float4 q = sQd[e];                        // ds_load_b128: a,b,ymn,ymx
      if (cy > q.z && cy < q.w) {
        float xi = (cy - q.y) / q.x;
        n++;
        xallmx = fmaxf(xallmx, xi);
        xallmn = fminf(xallmn, xi);
        if (xi >= cx) { xs = fminf(xs, xi); anyS = 1; }
        else          { xinf = fmaxf(xinf, xi); anyI = 1; }
      }
    }
#pragma unroll
    for (int off = 16; off; off >>= 1) {
      n      += __shfl_xor(n, off, 32);
      xallmx  = fmaxf(xallmx, __shfl_xor(xallmx, off, 32));
      xallmn  = fminf(xallmn, __shfl_xor(xallmn, off, 32));
      xs      = fminf(xs,     __shfl_xor(xs, off, 32));
      xinf    = fmaxf(xinf,   __shfl_xor(xinf, off, 32));
      anyS   |= __shfl_xor(anyS, off, 32);
      anyI   |= __shfl_xor(anyI, off, 32);
    }
    const bool valid = (n == 2) || ((n > 2) && anyS && anyI);
    const float dx = (n == 2) ? (xallmx - xallmn) : (xs - xinf);
    const float L2 = fabsf(dx);
    const float d1 = fabsf(cy - 1.0f);
    const float d2 = fabsf(px - 1.0f);

    // ---- table L1-min over T=512 rows (one b128 per row per lane) --------
    float pm = FLTMAXF;
    for (int t = lane; t < T_N; t += 32) {
      float4 r = sTr[t];
      float al = fabsf(L1 - r.x) + fabsf(L2 - r.y) +
                 fabsf(d1 - r.z) + fabsf(d2 - r.w);
      pm = fminf(pm, al);
    }
#pragma unroll
    for (int off = 16; off; off >>= 1)
      pm = fminf(pm, __shfl_xor(pm, off, 32));

    if (lane == 0) {
      mOut[pt] = valid ? pm : BIGF;
      vOut[pt] = valid ? 1 : 0;
    }
  }
}

// ---------------------------------------------------------------------------
// Kernel 3: cummin over P + masked sum. Async-stage to LDS, then a two-level
// block scan: 8 waves do the 128 independent chunk scans, one wave does the
// 128-wide exclusive prefix-min, all threads combine + masked sum.
// ---------------------------------------------------------------------------
__global__ void __launch_bounds__(SCAN_TPB)
scan_kernel(const float* __restrict__ m,
            const int*   __restrict__ valid,
            float* __restrict__ out) {
  __shared__ __align__(16) float sM[P_N];
  __shared__ __align__(16) int   sV[P_N];
  __shared__ float sCM[P_N / 32];      // chunk minima
  __shared__ float sPre[P_N / 32];     // exclusive prefix-min per chunk
  __shared__ float sSum[SCAN_TPB / 32];

  const int tid = threadIdx.x, lane = tid & 31, wave = tid >> 5;

  for (int i = tid; i < P_N / 4; i += SCAN_TPB) {
    async_copy_b128(m + 4 * i, sM + 4 * i);
    async_copy_b128((const float*)(valid + 4 * i), (float*)(sV + 4 * i));
  }
  async_wait_all();
  __syncthreads();

  // Chunk-local inclusive min-scans (independent, 8 waves x 16 chunks).
  for (int c = wave; c < P_N / 32; c += SCAN_TPB / 32) {
    float v = sM[c * 32 + lane];
#pragma unroll
    for (int off = 1; off < 32; off <<= 1) {
      float u = __shfl_up(v, off, 32);
      if (lane >= off) v = fminf(v, u);
    }
    sM[c * 32 + lane] = v;
    if (lane == 31) sCM[c] = v;
  }
  __syncthreads();

  // Exclusive prefix-min over 128 chunk minima (wave 0, 4 rounds).
  if (wave == 0) {
    float carry = FLTMAXF;
    for (int base = 0; base < P_N / 32; base += 32) {
      float inc = sCM[base + lane];
#pragma unroll
      for (int off = 1; off < 32; off <<= 1) {
        float u = __shfl_up(inc, off, 32);
        if (lane >= off) inc = fminf(inc, u);
      }
      float shifted = __shfl_up(inc, 1, 32);
      float excl = (lane == 0) ? carry : fminf(carry, shifted);
      sPre[base + lane] = excl;
      carry = fminf(carry, __shfl(inc, 31, 32));
    }
  }
  __syncthreads();

  // run[i] = min(chunkScan[i], chunkPrefix); masked sum.
  float acc = 0.0f;
  for (int i = tid; i < P_N; i += SCAN_TPB) {
    float run = fminf(sM[i], sPre[i >> 5]);
    if (sV[i]) acc += run;
  }
#pragma unroll
  for (int off = 16; off; off >>= 1)
    acc += __shfl_xor(acc, off, 32);
  if (lane == 0) sSum[wave] = acc;
  __syncthreads();
  if (tid == 0) {
    float s = 0.0f;
#pragma unroll
    for (int w = 0; w < SCAN_TPB / 32; ++w) s += sSum[w];
    out[0] = s;
  }
}

// ---------------------------------------------------------------------------
extern "C" void kernel_launch(void* const* d_in, const int* in_sizes, int n_in,
                              void* d_out, int out_size, void* d_ws, size_t ws_size,
                              hipStream_t stream) {
  const float* verts   = (const float*)d_in[0];  // (V,3)
  const float* tableR  = (const float*)d_in[1];  // (T,7)
  const int*   edges   = (const int*)d_in[2];    // (E,2)
  const int*   listAll = (const int*)d_in[3];    // (P,)
  // d_in[4]=epoch, d_in[5]=pth: unused by the loss.

  float* ep   = (float*)d_ws;        // 6*E floats  (96KB), packed pair+quad
  float* mArr = ep + 6 * E_N;        // P floats    (16KB)
  int*   vArr = (int*)(mArr + P_N);  // P ints      (16KB)

  edge_params_kernel<<<E_N / 256, 256, 0, stream>>>(verts, edges, ep);
  point_kernel<<<P_N / PTS_PER_BLOCK, TPB, 0, stream>>>(verts, tableR, listAll,
                                                        ep, mArr, vArr);
  scan_kernel<<<1, SCAN_TPB, 0, stream>>>(mArr, vArr, (float*)d_out);
}